// ApproximateLoss_60129542144623
// MI455X (gfx1250) — compile-verified
//
#include <hip/hip_runtime.h>
#include <math.h>

// Problem constants (from reference)
#define V_SIZE      50257
#define N_ROWS      2048
#define NUM_SAMP    250
#define ALPHA_F     0.75f

// ws layout (floats): [0, V_SIZE) cdf ; [50432, 50432+N_ROWS) per-row losses
#define CDF_PAD     50432

typedef __attribute__((ext_vector_type(2))) float v2f;
typedef __attribute__((ext_vector_type(8))) float v8f;

// ---------------------------------------------------------------------------
// Kernel 1: fused u = unigram^0.75 + single-workgroup inclusive scan -> cdf.
// 50257 elems / 1024 threads = 50 chunks of Hillis-Steele in LDS; microseconds.
// ---------------------------------------------------------------------------
__global__ __launch_bounds__(1024) void scan_kernel(const float* __restrict__ unigram,
                                                    float* __restrict__ cdf) {
    __shared__ float buf[2][1024];
    __shared__ float carry_s;
    const int tid = threadIdx.x;
    if (tid == 0) carry_s = 0.0f;
    __syncthreads();

    for (int base = 0; base < V_SIZE; base += 1024) {
        const int j = base + tid;
        float val = 0.0f;
        if (j < V_SIZE) {
            // unigram > 0 guaranteed; fast pow via exp/log
            val = __expf(ALPHA_F * __logf(unigram[j]));
        }
        int pin = 0;
        buf[0][tid] = val;
        __syncthreads();
        #pragma unroll
        for (int off = 1; off < 1024; off <<= 1) {
            float v = buf[pin][tid];
            if (tid >= off) v += buf[pin][tid - off];
            buf[pin ^ 1][tid] = v;
            pin ^= 1;
            __syncthreads();
        }
        const float incl = buf[pin][tid] + carry_s;
        if (j < V_SIZE) cdf[j] = incl;
        __syncthreads();
        if (tid == 1023) carry_s = incl;   // new running prefix
        __syncthreads();
    }
}

// first index with a[idx] >= v  (searchsorted 'left')
__device__ __forceinline__ int lower_bound_lds(const float* a, float v) {
    int lo = 0, hi = V_SIZE;
    while (lo < hi) {
        const int mid = (lo + hi) >> 1;
        if (a[mid] < v) lo = mid + 1; else hi = mid;
    }
    return lo;
}

// ---------------------------------------------------------------------------
// Kernel 2: per-row importance sampling + Z accumulation.
// Whole global CDF (196 KB) is staged in LDS (CDNA5 WGP has 320 KB) using the
// gfx1250 async memory->LDS path (GLOBAL_LOAD_ASYNC_TO_LDS_B128, ASYNCcnt):
// no VGPR round trip, no DScnt traffic. One wave32 per row; each lane handles
// ~8 of the 250 samples. Masked-row sampling = global-CDF search with a
// two-sided correction: v = u*(S - p_t); j = lb(v); if (j >= t) j = lb(v+p_t)
// ---------------------------------------------------------------------------
__global__ __launch_bounds__(256) void sample_kernel(const float* __restrict__ logits,
                                                     const int*   __restrict__ targets,
                                                     const float* __restrict__ cdf_g,
                                                     float*       __restrict__ row_loss) {
    __shared__ __align__(16) float s_cdf[V_SIZE];

    // Low 32 bits of the flat shared-aperture address == LDS byte offset.
    const unsigned lds_base = (unsigned)(uintptr_t)(&s_cdf[0]);

    // Main body: 50256 floats as 12564 x B128 async copies (16 B per lane).
    for (int j4 = threadIdx.x; j4 < (V_SIZE >> 2); j4 += blockDim.x) {
        const unsigned byte = (unsigned)j4 << 4;
        asm volatile("global_load_async_to_lds_b128 %0, %1, %2 offset:0"
                     :: "v"(lds_base + byte), "v"(byte), "s"(cdf_g)
                     : "memory");
    }
    // Tail: element 50256 as a single B32 async copy from lane 0.
    if (threadIdx.x == 0) {
        const unsigned byte = (unsigned)(V_SIZE & ~3) * 4u;
        asm volatile("global_load_async_to_lds_b32 %0, %1, %2 offset:0"
                     :: "v"(lds_base + byte), "v"(byte), "s"(cdf_g)
                     : "memory");
    }
    asm volatile("s_wait_asynccnt 0x0" ::: "memory");
    __syncthreads();

    const float S    = s_cdf[V_SIZE - 1];
    const int   lane = threadIdx.x & 31;
    const int   wave = threadIdx.x >> 5;
    const int   wglobal = blockIdx.x * (blockDim.x >> 5) + wave;
    const int   nwaves  = gridDim.x * (blockDim.x >> 5);

    for (int r = wglobal; r < N_ROWS; r += nwaves) {
        const int t = targets[r];
        const float* lrow = logits + (size_t)r * V_SIZE;
        __builtin_prefetch(lrow + t, 0, 1);           // global_prefetch_b8
        const float tl   = lrow[t];
        const float ctm1 = (t > 0) ? s_cdf[t - 1] : 0.0f;
        const float pt   = s_cdf[t] - ctm1;           // raw unigram mass of target
        const float mtot = S - pt;                    // masked-row total mass

        float acc = 0.0f;
        for (int s = lane; s < NUM_SAMP; s += 32) {
            // deterministic per-(row,sample) hash -> u in [0,1)
            unsigned h = (unsigned)(r * NUM_SAMP + s) * 2654435761u + 0x9E3779B9u;
            h ^= h >> 16; h *= 0x7feb352du;
            h ^= h >> 15; h *= 0x846ca68bu;
            h ^= h >> 16;
            const float u = (float)(h >> 8) * (1.0f / 16777216.0f);
            const float v = u * mtot;

            int j = lower_bound_lds(s_cdf, v);
            if (j >= t) j = lower_bound_lds(s_cdf, v + pt);   // skip over masked target
            if (j > V_SIZE - 1) j = V_SIZE - 1;

            const float uraw = s_cdf[j] - ((j > 0) ? s_cdf[j - 1] : 0.0f);
            const float w    = S / ((float)NUM_SAMP * uraw);  // 1/(S * uni[j]) scaling
            acc += w * __expf(lrow[j]);
        }
        // wave32 reduction
        #pragma unroll
        for (int off = 16; off > 0; off >>= 1) acc += __shfl_xor(acc, off, 32);

        if (lane == 0) {
            const float Z = __expf(tl) + acc;
            row_loss[r] = -(tl - __logf(Z));
        }
    }
}

// ---------------------------------------------------------------------------
// Kernel 3: mean over 2048 row losses using the matrix pipe.
// A = ones(16x4)  =>  D[m,n] = sum_k B[k,n] + C[m,n]; 32 chained WMMAs
// reduce 2048 floats (B element order is irrelevant for the grand total),
// then lanes 0..15 of row M=0 are summed by shuffles. One wave32, EXEC all 1s.
// ---------------------------------------------------------------------------
__global__ __launch_bounds__(32) void reduce_kernel(const float* __restrict__ row_loss,
                                                    float* __restrict__ out) {
    const int lane = threadIdx.x;
    v8f c = {};
    v2f a; a[0] = 1.0f; a[1] = 1.0f;      // all-ones A covers the full 16x4 tile

    #pragma unroll 4
    for (int chunk = 0; chunk < N_ROWS / 64; ++chunk) {
        // contiguous pair per lane -> single global_load_b64
        v2f b = *reinterpret_cast<const v2f*>(row_loss + chunk * 64 + lane * 2);
        c = __builtin_amdgcn_wmma_f32_16x16x4_f32(
                /*neg_a=*/false, a, /*neg_b=*/false, b,
                /*c_mod=*/(short)0, c, /*reuse_a=*/false, /*reuse_b=*/false);
    }
    // D row M=0 lives in VGPR0 of lanes 0..15 (N = lane); lanes 16..31 duplicate row M=8.
    float partial = (lane < 16) ? c[0] : 0.0f;
    #pragma unroll
    for (int off = 16; off > 0; off >>= 1) partial += __shfl_xor(partial, off, 32);
    if (lane == 0) out[0] = partial * (1.0f / (float)N_ROWS);
}

// ---------------------------------------------------------------------------
extern "C" void kernel_launch(void* const* d_in, const int* in_sizes, int n_in,
                              void* d_out, int out_size, void* d_ws, size_t ws_size,
                              hipStream_t stream) {
    const float* logits  = (const float*)d_in[0];   // [2048, 50257] f32
    const float* unigram = (const float*)d_in[1];   // [50257] f32
    const int*   targets = (const int*)d_in[2];     // [2048] int
    float* out = (float*)d_out;                     // scalar f32
    float* ws  = (float*)d_ws;
    float* cdf  = ws;                               // V_SIZE floats
    float* rows = ws + CDF_PAD;                     // N_ROWS floats

    scan_kernel<<<1, 1024, 0, stream>>>(unigram, cdf);
    sample_kernel<<<64, 256, 0, stream>>>(logits, targets, cdf, rows);
    reduce_kernel<<<1, 32, 0, stream>>>(rows, out);
}